// SymPBC_54451595378999
// MI455X (gfx1250) — compile-verified
//
#include <hip/hip_runtime.h>

#ifndef __has_builtin
#define __has_builtin(x) 0
#endif

// ---------------- problem constants (match reference setup) ----------------
constexpr int kW    = 16384;  // window length
constexpr int kB    = 2;      // batch
constexpr int kS    = 118;    // |index set| for L=50, rho=1
constexpr int kSP   = 128;    // padded to 4 chunks of 32 lanes
constexpr int kTW   = 64;     // w-values per block
constexpr int kHALO = 32;     // max |shift| is 26 -> 32 halo each side
constexpr int kNPOS = kTW + 2 * kHALO;   // 128 staged positions
constexpr int kWPW  = kTW / 8;           // w per wave (8 waves / 256 thr)

typedef __attribute__((ext_vector_type(2))) float v2f;
typedef int v2i __attribute__((vector_size(2 * sizeof(int))));

// ---------------- compile-time (m,n,w1,w2) table, reference order ----------
struct STab { int m[kSP]; int n[kSP]; float w1[kSP]; float w2[kSP]; };

constexpr STab makeTab() {
    STab t{};
    int c = 0;
    for (int m = -25; m <= 25; ++m) {
        for (int n = -25; n <= 25; ++n) {
            const int am = m < 0 ? -m : m;
            const int p  = m * n;
            const int ap = p < 0 ? -p : p;
            if (n >= am && ap <= 25) {          // |mn| <= rho*L//2, n >= |m|
                t.m[c]  = m;
                t.n[c]  = n;
                t.w1[c] = (m == 0 && n == 0) ? 0.5f : 1.0f;
                t.w2[c] = (n > am) ? 1.0f : 0.0f;
                ++c;
            }
        }
    }
    for (; c < kSP; ++c) { t.m[c] = 0; t.n[c] = 0; t.w1[c] = 0.0f; t.w2[c] = 0.0f; }
    return t;
}

__constant__ STab g_tab = makeTab();

// dotc(X,Y) = sum over 2 modes of X[p] * conj(Y[p]); layout {re0,re1,im0,im1}
__device__ __forceinline__ v2f dotc(const float4 X, const float4 Y) {
    v2f r;
    r.x = X.x * Y.x + X.z * Y.z + X.y * Y.y + X.w * Y.w;   // real
    r.y = X.z * Y.x - X.x * Y.z + X.w * Y.y - X.y * Y.w;   // imag
    return r;
}

__global__ __launch_bounds__(256)
void sympbc_kernel(const float* __restrict__ Er,
                   const float* __restrict__ Ei,
                   float* __restrict__ out)
{
    __shared__ float s_tile[kNPOS * 4] __attribute__((aligned(16)));

    const int b  = blockIdx.y;
    const int w0 = blockIdx.x * kTW;
    const int t  = threadIdx.x;

    // -------- stage tile + halo into LDS as {re0,re1,im0,im1} per w --------
    {
        const int p    = t >> 1;      // staged position 0..127
        const int isim = t & 1;       // 0 = reals pair, 1 = imags pair
        const int wp   = (w0 - kHALO + p) & (kW - 1);
        const float* src = (isim ? Ei : Er) + ((size_t)b * kW + wp) * 2;
        float*       dst = &s_tile[p * 4 + isim * 2];
#if defined(__gfx1250__) && __has_builtin(__builtin_amdgcn_global_load_async_to_lds_b64)
        __builtin_amdgcn_global_load_async_to_lds_b64(
            (__attribute__((address_space(1))) v2i*)(v2i*)const_cast<float*>(src),
            (__attribute__((address_space(3))) v2i*)(v2i*)dst,
            0, 0);
        asm volatile("s_wait_asynccnt 0" ::: "memory");
#else
        const float rlo = src[0], rhi = src[1];
        dst[0] = rlo; dst[1] = rhi;
#endif
    }
    __syncthreads();

    // -------- per-lane table registers: lane covers s = c*32 + lane --------
    const int wave = t >> 5;
    const int lane = t & 31;
    int   tm[4], tn[4];
    float tw1[4], tw2[4];
#pragma unroll
    for (int c = 0; c < 4; ++c) {
        const int s = c * 32 + lane;
        tm[c]  = g_tab.m[s];
        tn[c]  = g_tab.n[s];
        tw1[c] = g_tab.w1[s];
        tw2[c] = g_tab.w2[s];
    }

    const size_t wstride = (size_t)2 * kS * 2;  // 472 floats per w  [M,S,2]
    float* outb = out + ((size_t)b * kW + w0) * wstride;

#pragma unroll 2
    for (int i = 0; i < kWPW; ++i) {
        const int wl   = kHALO + wave * kWPW + i;              // local w
        float*    outw = outb + (size_t)(wave * kWPW + i) * wstride;

#pragma unroll
        for (int c = 0; c < 4; ++c) {
            const int m = tm[c], n = tn[c];
            const float4 A  = *(const float4*)&s_tile[(wl - m)     * 4]; // E[w-m]   = Rm
            const float4 Bv = *(const float4*)&s_tile[(wl - n)     * 4]; // E[w-n]   = Rn
            const float4 C  = *(const float4*)&s_tile[(wl - m - n) * 4]; // E[w-m-n] = Rmn
            const float4 D  = *(const float4*)&s_tile[(wl + m)     * 4]; // E[w+m]   = R_m
            const float4 G  = *(const float4*)&s_tile[(wl + n)     * 4]; // E[w+n]   = R_n
            const float4 H  = *(const float4*)&s_tile[(wl + m + n) * 4]; // E[w+m+n] = R_mn

            const v2f S1  = dotc(Bv, C);   // sum Rn  conj(Rmn)
            const v2f S1b = dotc(G,  H);   // sum R_n conj(R_mn)
            const v2f S2  = dotc(A,  C);   // sum Rm  conj(Rmn)
            const v2f S2b = dotc(D,  H);   // sum R_m conj(R_mn)

            // T1 = S1*Rm + S1b*R_m ; T2 = S2*Rn + S2b*R_n   (complex, per mode)
            const float t1r0 = S1.x*A.x  - S1.y*A.z  + S1b.x*D.x - S1b.y*D.z;
            const float t1i0 = S1.x*A.z  + S1.y*A.x  + S1b.x*D.z + S1b.y*D.x;
            const float t1r1 = S1.x*A.y  - S1.y*A.w  + S1b.x*D.y - S1b.y*D.w;
            const float t1i1 = S1.x*A.w  + S1.y*A.y  + S1b.x*D.w + S1b.y*D.y;

            const float t2r0 = S2.x*Bv.x - S2.y*Bv.z + S2b.x*G.x - S2b.y*G.z;
            const float t2i0 = S2.x*Bv.z + S2.y*Bv.x + S2b.x*G.z + S2b.y*G.x;
            const float t2r1 = S2.x*Bv.y - S2.y*Bv.w + S2b.x*G.y - S2b.y*G.w;
            const float t2i1 = S2.x*Bv.w + S2.y*Bv.y + S2b.x*G.w + S2b.y*G.y;

            const float w1 = tw1[c], w2 = tw2[c];
            v2f f0, f1;
            f0.x = w1 * t1r0 + w2 * t2r0;  f0.y = w1 * t1i0 + w2 * t2i0;
            f1.x = w1 * t1r1 + w2 * t2r1;  f1.y = w1 * t1i1 + w2 * t2i1;

            const int s = c * 32 + lane;
            if (s < kS) {
                // [B,W,M,S,2]: mode0 at s*2, mode1 at 2*kS + s*2 (coalesced v2f)
                __builtin_nontemporal_store(f0, (v2f*)(outw + 2 * s));
                __builtin_nontemporal_store(f1, (v2f*)(outw + 2 * kS + 2 * s));
            }
        }
    }
}

extern "C" void kernel_launch(void* const* d_in, const int* in_sizes, int n_in,
                              void* d_out, int out_size, void* d_ws, size_t ws_size,
                              hipStream_t stream) {
    const float* Er  = (const float*)d_in[0];
    const float* Ei  = (const float*)d_in[1];
    float*       out = (float*)d_out;
    (void)in_sizes; (void)n_in; (void)out_size; (void)d_ws; (void)ws_size;

    dim3 grid(kW / kTW, kB);   // 256 tiles x 2 batches = 512 blocks
    dim3 block(256);           // 8 wave32
    sympbc_kernel<<<grid, block, 0, stream>>>(Er, Ei, out);
}